// EmbeddingDropout_29875792511459
// MI455X (gfx1250) — compile-verified
//
#include <hip/hip_runtime.h>
#include <cstdint>

// EmbeddingDropout gather for MI455X (gfx1250).
// Pure bandwidth problem: ~64MiB gathered reads + 64MiB streamed writes,
// ~5.5us at 23.3 TB/s. No FLOPs -> no WMMA; the CDNA5 win is the async
// global->LDS DMA path (ASYNCcnt) + b128 accesses + NT stores, with the
// threefry dropout mask computed entirely in SALU under the DMA shadow.

#define VOCAB 50257u
#define DIM   1024u
#define HALF  25129u   // ceil(50257/2): JAX threefry pads odd count with one 0

typedef float v4f __attribute__((ext_vector_type(4)));  // native vec: OK for NT builtins

__device__ __forceinline__ uint32_t rotl32(uint32_t x, int d) {
    return (x << d) | (x >> (32 - d));
}

// Exact JAX threefry2x32 replication for
//   keep = jax.random.bernoulli(jax.random.key(42), 0.9, (VOCAB, 1))
// key data = {0, 42}; counters = iota(VOCAB) padded to even, split in half:
//   x0[j] = j                      (j in [0, HALF))
//   x1[j] = HALF + j  (or 0 pad when HALF + j == VOCAB)
// bits[row] = x0-lane output if row < HALF else x1-lane output (j = row-HALF).
__device__ __forceinline__ float row_scale(uint32_t row) {
    const uint32_t j    = (row < HALF) ? row : (row - HALF);
    const uint32_t lane = (row < HALF) ? 0u  : 1u;

    uint32_t x0 = j;
    uint32_t x1 = (j + HALF < VOCAB) ? (j + HALF) : 0u;  // padded zero counter

    const uint32_t ks0 = 0u;
    const uint32_t ks1 = 42u;
    const uint32_t ks2 = 0x1BD11BDAu ^ 0u ^ 42u;

    x0 += ks0; x1 += ks1;
#define QR(r) { x0 += x1; x1 = rotl32(x1, r); x1 ^= x0; }
    QR(13) QR(15) QR(26) QR(6)  x0 += ks1; x1 += ks2 + 1u;   // i=0
    QR(17) QR(29) QR(16) QR(24) x0 += ks2; x1 += ks0 + 2u;   // i=1
    QR(13) QR(15) QR(26) QR(6)  x0 += ks0; x1 += ks1 + 3u;   // i=2
    QR(17) QR(29) QR(16) QR(24) x0 += ks1; x1 += ks2 + 4u;   // i=3
    QR(13) QR(15) QR(26) QR(6)  x0 += ks2; x1 += ks0 + 5u;   // i=4
#undef QR

    const uint32_t bits = lane ? x1 : x0;
    // jax.random.uniform: bitcast((bits>>9)|0x3f800000) - 1.0, then (u < p)
    const float u = __uint_as_float((bits >> 9) | 0x3f800000u) - 1.0f;
    return (u < 0.9f) ? (1.0f / 0.9f) : 0.0f;
}

// One 256-thread block (8 wave32) per token: each lane moves one 16B chunk
// of the 4KB embedding row.
__global__ __launch_bounds__(256) void emb_dropout_gather(
    const unsigned long long* __restrict__ x,   // [B*S] int64 token ids
    const float* __restrict__ weight,           // [VOCAB, DIM] fp32
    float* __restrict__ out)                    // [B*S, DIM] fp32
{
    __shared__ __align__(16) float lds_row[DIM];   // 4KB staging buffer

    const uint32_t token = blockIdx.x;
    const uint32_t tid   = threadIdx.x;
    const uint32_t row   = (uint32_t)x[token];     // block-uniform -> s_load

    // ---- Async DMA: weight row -> LDS, one b128 per lane (ASYNCcnt path) ----
    // Low 32 bits of a generic pointer to __shared__ == LDS byte offset.
    const uint32_t lds_off = (uint32_t)(uintptr_t)(&lds_row[0]) + tid * 16u;
    const uint64_t gsrc = (uint64_t)(uintptr_t)(weight + (size_t)row * DIM + (size_t)tid * 4u);
    asm volatile("global_load_async_to_lds_b128 %0, %1, off"
                 :: "v"(lds_off), "v"(gsrc)
                 : "memory");

    // Threefry dropout scale: pure SALU, overlapped with the in-flight DMA.
    const float s = row_scale(row);

    // Wait for this wave's async copy. Passing `s` as a dummy operand makes
    // the wait data-dependent on the threefry result, forcing the compiler to
    // schedule all of row_scale() BEFORE the stall (true DMA/ALU overlap).
    asm volatile("s_wait_asynccnt 0" :: "s"(__float_as_uint(s)) : "memory");
    __syncthreads();

    // ---- Scale and stream out (non-temporal b128: write-once output) ----
    v4f v = *(const v4f*)(&lds_row[tid * 4u]);
    v *= s;
    v4f* dst = (v4f*)(out + (size_t)token * DIM + (size_t)tid * 4u);
    __builtin_nontemporal_store(v, dst);
}

extern "C" void kernel_launch(void* const* d_in, const int* in_sizes, int n_in,
                              void* d_out, int out_size, void* d_ws, size_t ws_size,
                              hipStream_t stream) {
    const unsigned long long* x      = (const unsigned long long*)d_in[0]; // int64 ids
    const float*              weight = (const float*)d_in[1];              // fp32 [V,D]
    float*                    out    = (float*)d_out;

    const int tokens = in_sizes[0];           // 8 * 2048 = 16384
    emb_dropout_gather<<<dim3(tokens), dim3(256), 0, stream>>>(x, weight, out);
}